// RIConv2SetAbstraction_5282809774328
// MI455X (gfx1250) — compile-verified
//
#include <hip/hip_runtime.h>
#include <hip/hip_bf16.h>
#include <math.h>

#define BB 8
#define NN 8192
#define SS 2048
#define KK 32
#define CP 64
#define NPIX (BB*KK*SS)   // 524288
#define EPS 1e-7f

typedef __attribute__((ext_vector_type(2))) float v2f;
typedef __attribute__((ext_vector_type(8))) float v8f;

// ---------------------------------------------------------------------------
// 1) Furthest point sampling: one workgroup per batch, serial over S samples.
// ---------------------------------------------------------------------------
#define FPS_T 512
__global__ void fps_kernel(const float* __restrict__ xyz, int* __restrict__ fps_idx) {
    __shared__ float sv[FPS_T];
    __shared__ int   si[FPS_T];
    __shared__ float fpt[3];
    const int b = blockIdx.x;
    const int t = threadIdx.x;
    const float* px = xyz + (size_t)b * NN * 3;
    const int PPT = NN / FPS_T;  // 16 points per thread
    float mx[NN / FPS_T], my[NN / FPS_T], mz[NN / FPS_T], dmin[NN / FPS_T];
    for (int c = 0; c < PPT; ++c) {
        int j = t + c * FPS_T;
        mx[c] = px[j*3+0]; my[c] = px[j*3+1]; mz[c] = px[j*3+2];
        dmin[c] = 1e10f;
    }
    int far = 0;
    for (int i = 0; i < SS; ++i) {
        if (t == 0) {
            fps_idx[b * SS + i] = far;
            fpt[0] = px[far*3+0]; fpt[1] = px[far*3+1]; fpt[2] = px[far*3+2];
        }
        __syncthreads();
        float fx = fpt[0], fy = fpt[1], fz = fpt[2];
        float best = -1.0f; int bj = 0;
        for (int c = 0; c < PPT; ++c) {
            float dx = mx[c]-fx, dy = my[c]-fy, dz = mz[c]-fz;
            float d = dx*dx + dy*dy + dz*dz;
            dmin[c] = fminf(dmin[c], d);
            if (dmin[c] > best) { best = dmin[c]; bj = t + c*FPS_T; }
        }
        sv[t] = best; si[t] = bj;
        __syncthreads();
        for (int stride = FPS_T/2; stride > 0; stride >>= 1) {
            if (t < stride) {
                float ov = sv[t+stride]; int oi = si[t+stride];
                if (ov > sv[t] || (ov == sv[t] && oi < si[t])) { sv[t] = ov; si[t] = oi; }
            }
            __syncthreads();
        }
        far = si[0];
        __syncthreads();
    }
}

// ---------------------------------------------------------------------------
// 2) Gather sampled points/normals straight into the output buffer.
// ---------------------------------------------------------------------------
__global__ void gather_new(const float* __restrict__ xyz, const float* __restrict__ norm,
                           const int* __restrict__ fps_idx,
                           float* __restrict__ out_xyz, float* __restrict__ out_norm) {
    int i = blockIdx.x * blockDim.x + threadIdx.x;
    if (i >= BB * SS) return;
    int b = i / SS;
    int n = fps_idx[i];
    size_t src = ((size_t)b * NN + n) * 3;
    out_xyz[i*3+0]  = xyz[src+0];  out_xyz[i*3+1]  = xyz[src+1];  out_xyz[i*3+2]  = xyz[src+2];
    out_norm[i*3+0] = norm[src+0]; out_norm[i*3+1] = norm[src+1]; out_norm[i*3+2] = norm[src+2];
}

// ---------------------------------------------------------------------------
// 3) KNN: one block per query. All N distances in LDS, 32 argmin passes
//    (low-index tie-break == stable top_k).
// ---------------------------------------------------------------------------
#define KNN_T 256
__global__ void knn_kernel(const float* __restrict__ xyz, const float* __restrict__ new_xyz,
                           int* __restrict__ knn_idx) {
    __shared__ float dist[NN];     // 32 KB
    __shared__ float rv[KNN_T];
    __shared__ int   rix[KNN_T];
    const int q = blockIdx.x;      // b*S + s
    const int b = q / SS;
    const int t = threadIdx.x;
    float qx = new_xyz[q*3+0], qy = new_xyz[q*3+1], qz = new_xyz[q*3+2];
    float qq = qx*qx + qy*qy + qz*qz;
    const float* px = xyz + (size_t)b * NN * 3;
    for (int j = t; j < NN; j += KNN_T) {
        float x = px[j*3+0], y = px[j*3+1], z = px[j*3+2];
        dist[j] = qq + (x*x + y*y + z*z) - 2.0f*(qx*x + qy*y + qz*z);
    }
    __syncthreads();
    for (int kk = 0; kk < KK; ++kk) {
        float best = 3.4e38f; int bj = NN;
        for (int j = t; j < NN; j += KNN_T)
            if (dist[j] < best) { best = dist[j]; bj = j; }
        rv[t] = best; rix[t] = bj;
        __syncthreads();
        for (int stride = KNN_T/2; stride > 0; stride >>= 1) {
            if (t < stride) {
                float ov = rv[t+stride]; int oi = rix[t+stride];
                if (ov < rv[t] || (ov == rv[t] && oi < rix[t])) { rv[t] = ov; rix[t] = oi; }
            }
            __syncthreads();
        }
        if (t == 0) { knn_idx[(size_t)q*KK + kk] = rix[0]; dist[rix[0]] = 3.4e38f; }
        __syncthreads();
    }
}

// ---------------------------------------------------------------------------
// 4) RI features: one wave32 per (b,s). lane = neighbor k.
// ---------------------------------------------------------------------------
__device__ __forceinline__ float clip1(float x) {
    return fminf(fmaxf(x, -1.0f + EPS), 1.0f - EPS);
}

__global__ void ri_kernel(const float* __restrict__ xyz, const float* __restrict__ norm,
                          const float* __restrict__ new_xyz, const float* __restrict__ new_norm,
                          const int* __restrict__ knn_idx,
                          int* __restrict__ idx_ordered, float* __restrict__ x0) {
    const int lane = threadIdx.x & 31;
    const int wid  = blockIdx.x * (blockDim.x >> 5) + (threadIdx.x >> 5);
    if (wid >= BB * SS) return;
    const int b = wid / SS, s = wid % SS;

    float cx = new_xyz[wid*3+0], cy = new_xyz[wid*3+1], cz = new_xyz[wid*3+2];
    float nx = new_norm[wid*3+0], ny = new_norm[wid*3+1], nz = new_norm[wid*3+2];

    int nidx = knn_idx[(size_t)wid*KK + lane];
    size_t gsrc = ((size_t)b * NN + nidx) * 3;
    float lx = xyz[gsrc+0]-cx, ly = xyz[gsrc+1]-cy, lz = xyz[gsrc+2]-cz;

    // projection onto tangent plane of the query normal
    float dp = lx*nx + ly*ny + lz*nz;
    float ppx = lx - dp*nx, ppy = ly - dp*ny, ppz = lz - dp*nz;
    float plen = sqrtf(ppx*ppx + ppy*ppy + ppz*ppz);
    float inv = plen > 0.0f ? 1.0f/plen : 0.0f;
    float ux = ppx*inv, uy = ppy*inv, uz = ppz*inv;

    // wave argmax of plen (first-index tie-break)
    float mv = plen; int mi = lane;
    #pragma unroll
    for (int off = 16; off > 0; off >>= 1) {
        float ov = __shfl_xor(mv, off);
        int   oi = __shfl_xor(mi, off);
        if (ov > mv || (ov == mv && oi < mi)) { mv = ov; mi = oi; }
    }
    float vrx = __shfl(ux, mi), vry = __shfl(uy, mi), vrz = __shfl(uz, mi);

    float dots = ux*vrx + uy*vry + uz*vrz;
    float crx = uy*vrz - uz*vry, cry = uz*vrx - ux*vrz, crz = ux*vry - uy*vrx;
    float sg = crx*nx + cry*ny + crz*nz;
    float sgn = (sg > 0.0f) ? 1.0f : ((sg < 0.0f) ? -1.0f : 0.0f);
    if (lane == 0) sgn = 1.0f;
    float dval = sgn*dots - (1.0f - sgn);

    // bitonic sort 32 lanes: descending dval, ties -> ascending original k
    float sd = dval; int sk = lane;
    for (int size = 2; size <= 32; size <<= 1) {
        for (int stride = size >> 1; stride > 0; stride >>= 1) {
            float od = __shfl_xor(sd, stride);
            int   ok = __shfl_xor(sk, stride);
            bool up    = ((lane & size) == 0);
            bool lower = ((lane & stride) == 0);
            bool selfLess = (sd > od) || (sd == od && sk < ok);
            bool keep = up ? (lower ? selfLess : !selfLess)
                           : (lower ? !selfLess : selfLess);
            if (!keep) { sd = od; sk = ok; }
        }
    }

    int n_ord = __shfl(nidx, sk);
    idx_ordered[(size_t)wid*KK + lane] = n_ord;

    // recompute with ordered neighbors
    size_t osrc = ((size_t)b * NN + n_ord) * 3;
    float l2x = xyz[osrc+0]-cx, l2y = xyz[osrc+1]-cy, l2z = xyz[osrc+2]-cz;
    float glen = sqrtf(l2x*l2x + l2y*l2y + l2z*l2z);
    float ginv = glen > 0.0f ? 1.0f/glen : 0.0f;
    float u2x = l2x*ginv, u2y = l2y*ginv, u2z = l2z*ginv;
    float gnx = norm[osrc+0], gny = norm[osrc+1], gnz = norm[osrc+2];

    int pl = (lane + 31) & 31;   // previous element (roll by 1)
    float plx = __shfl(l2x, pl), ply = __shfl(l2y, pl), plz = __shfl(l2z, pl);
    float pgx = __shfl(gnx, pl), pgy = __shfl(gny, pl), pgz = __shfl(gnz, pl);
    float pd  = __shfl(sd, pl);
    float dlast = __shfl(sd, 31);

    float inx = l2x - plx, iny = l2y - ply, inz = l2z - plz;
    float ilen = sqrtf(inx*inx + iny*iny + inz*inz);
    float iinv = ilen > 0.0f ? 1.0f/ilen : 0.0f;
    float iux = inx*iinv, iuy = iny*iinv, iuz = inz*iinv;

    float a0 = u2x*nx + u2y*ny + u2z*nz;
    float a1 = u2x*gnx + u2y*gny + u2z*gnz;
    float an = acosf(clip1(gnx*nx + gny*ny + gnz*nz));
    an = (a0 < a1 ? 1.0f : -1.0f) * an;

    float i0 = iux*gnx + iuy*gny + iuz*gnz;
    float i1 = iux*pgx + iuy*pgy + iuz*pgz;
    float i2 = acosf(clip1(gnx*pgx + gny*pgy + gnz*pgz));
    i2 = (i0 < i1 ? 1.0f : -1.0f) * i2;

    float pia = (lane == 0) ? (-3.0f - dlast) : (sd - pd);

    size_t p = (size_t)b*KK*SS + (size_t)lane*SS + s;  // pixel index (b,k,s)
    x0[0*(size_t)NPIX + p] = glen;
    x0[1*(size_t)NPIX + p] = pia;
    x0[2*(size_t)NPIX + p] = a0;
    x0[3*(size_t)NPIX + p] = a1;
    x0[4*(size_t)NPIX + p] = an;
    x0[5*(size_t)NPIX + p] = i0;
    x0[6*(size_t)NPIX + p] = i1;
    x0[7*(size_t)NPIX + p] = i2;
}

// ---------------------------------------------------------------------------
// 5) Gather point features into rows 64..127 of the layer-2 input.
// ---------------------------------------------------------------------------
__global__ void gather_gp(const float* __restrict__ points, const int* __restrict__ idx_ordered,
                          float* __restrict__ dst /* = x2 + 64*NPIX */) {
    int t = blockIdx.x * blockDim.x + threadIdx.x;
    const int total = CP * NPIX;
    if (t >= total) return;
    int cp  = t / NPIX;
    int p   = t % NPIX;
    int b   = p / (KK*SS);
    int rem = p & (KK*SS - 1);
    int k   = rem / SS;
    int s   = rem & (SS - 1);
    int ni  = idx_ordered[((size_t)b*SS + s)*KK + k];
    dst[(size_t)cp * NPIX + p] = points[((size_t)b*CP + cp)*NN + ni];
}

// ---------------------------------------------------------------------------
// 6) f32 WMMA GEMM + bias + BN + ReLU.  out[o][p] = sum_c w[o][c]*x[c][p]
//    Register-blocked over M: one wave owns a 16-pixel column strip and ALL
//    COUT/16 M-tiles, so each activation (B) fragment is loaded exactly once
//    and feeds COUT/16 V_WMMA_F32_16X16X4_F32 ops. Weight (A) fragments are
//    tiny and stay L0-resident. MODE 0: store; MODE 1: atomicMax (max over K).
// ---------------------------------------------------------------------------
template<int CIN, int COUT, int MODE>
__global__ void gemm_bn_relu(const float* __restrict__ xin, const float* __restrict__ w,
                             const float* __restrict__ bias, const float* __restrict__ gamma,
                             const float* __restrict__ beta, const float* __restrict__ mean,
                             const float* __restrict__ var,
                             float* __restrict__ xout, float* __restrict__ outmax) {
    constexpr int MT = COUT / 16;
    const int lane = threadIdx.x & 31;
    const int wv   = blockIdx.x * (blockDim.x >> 5) + (threadIdx.x >> 5);
    const int n0   = wv * 16;             // column strip base
    if (n0 >= NPIX) return;

    const int rowa = (lane & 15);         // A: M = lane&15 (within tile)
    const int ksel = (lane >> 4) * 2;     // A/B: lanes 16-31 carry K+2,K+3
    const int col  = n0 + (lane & 15);    // B/D: N = lane&15

    v8f acc[MT];
    #pragma unroll
    for (int mt = 0; mt < MT; ++mt) acc[mt] = (v8f){};

    #pragma unroll
    for (int k = 0; k < CIN; k += 4) {
        v2f bfrag;
        bfrag.x = xin[(size_t)(k + ksel)     * NPIX + col];
        bfrag.y = xin[(size_t)(k + ksel + 1) * NPIX + col];
        #pragma unroll
        for (int mt = 0; mt < MT; ++mt) {
            v2f a;
            a.x = w[(mt*16 + rowa) * CIN + k + ksel];
            a.y = w[(mt*16 + rowa) * CIN + k + ksel + 1];
            acc[mt] = __builtin_amdgcn_wmma_f32_16x16x4_f32(false, a, false, bfrag,
                                                            (short)0, acc[mt], false, false);
        }
    }

    #pragma unroll
    for (int mt = 0; mt < MT; ++mt) {
        const int obase = mt*16 + ((lane >> 4) * 8);  // D: M = r + 8*(lane>=16)
        #pragma unroll
        for (int r = 0; r < 8; ++r) {
            int o = obase + r;
            float scale = gamma[o] * rsqrtf(var[o] + 1e-5f);
            float val = (acc[mt][r] + bias[o] - mean[o]) * scale + beta[o];
            val = fmaxf(val, 0.0f);
            if (MODE == 0) {
                xout[(size_t)o * NPIX + col] = val;
            } else {
                int b   = col / (KK*SS);
                int rem = col & (KK*SS - 1);
                int s   = rem & (SS - 1);
                atomicMax(&outmax[((size_t)b * 128 + o) * SS + s], val);
            }
        }
    }
}

__global__ void zero_kernel(float* __restrict__ p, int n) {
    int i = blockIdx.x * blockDim.x + threadIdx.x;
    if (i < n) p[i] = 0.0f;
}

// ---------------------------------------------------------------------------
extern "C" void kernel_launch(void* const* d_in, const int* in_sizes, int n_in,
                              void* d_out, int out_size, void* d_ws, size_t ws_size,
                              hipStream_t stream) {
    const float* xyz    = (const float*)d_in[0];
    const float* norm   = (const float*)d_in[1];
    const float* points = (const float*)d_in[2];
    const float* w0 = (const float*)d_in[3];
    const float* b0 = (const float*)d_in[4];
    const float* g0 = (const float*)d_in[5];
    const float* t0 = (const float*)d_in[6];
    const float* m0 = (const float*)d_in[7];
    const float* v0 = (const float*)d_in[8];
    const float* w1 = (const float*)d_in[9];
    const float* b1 = (const float*)d_in[10];
    const float* g1 = (const float*)d_in[11];
    const float* t1 = (const float*)d_in[12];
    const float* m1 = (const float*)d_in[13];
    const float* v1 = (const float*)d_in[14];
    const float* w2 = (const float*)d_in[15];
    const float* b2 = (const float*)d_in[16];
    const float* g2 = (const float*)d_in[17];
    const float* t2 = (const float*)d_in[18];
    const float* m2 = (const float*)d_in[19];
    const float* v2 = (const float*)d_in[20];

    float* out      = (float*)d_out;
    float* out_xyz  = out;                       // [B,S,3]
    float* out_norm = out + (size_t)BB*SS*3;     // [B,S,3]
    float* out_pts  = out + (size_t)2*BB*SS*3;   // [B,128,S]

    char* ws = (char*)d_ws;
    size_t off = 0;
    int*   fps_idx = (int*)(ws + off);   off += (size_t)BB*SS*sizeof(int);
    int*   knn_idx = (int*)(ws + off);   off += (size_t)BB*SS*KK*sizeof(int);
    int*   idx_ord = (int*)(ws + off);   off += (size_t)BB*SS*KK*sizeof(int);
    float* x0      = (float*)(ws + off); off += (size_t)8   * NPIX * sizeof(float);
    float* x1      = (float*)(ws + off); off += (size_t)32  * NPIX * sizeof(float);
    float* x2      = (float*)(ws + off); off += (size_t)128 * NPIX * sizeof(float);

    // Stage 1: FPS + gather
    fps_kernel<<<BB, FPS_T, 0, stream>>>(xyz, fps_idx);
    gather_new<<<(BB*SS + 255)/256, 256, 0, stream>>>(xyz, norm, fps_idx, out_xyz, out_norm);

    // Stage 2: KNN
    knn_kernel<<<BB*SS, KNN_T, 0, stream>>>(xyz, out_xyz, knn_idx);

    // Stage 3: rotation-invariant features (one wave per query)
    ri_kernel<<<(BB*SS)/8, 256, 0, stream>>>(xyz, norm, out_xyz, out_norm, knn_idx, idx_ord, x0);

    // Stage 4: gather point features into x2 rows 64..127
    gather_gp<<<(CP*NPIX + 255)/256, 256, 0, stream>>>(points, idx_ord, x2 + (size_t)64*NPIX);

    // Stage 5: conv chain as f32 WMMA GEMMs (one 16-col strip per wave,
    // register-blocked over all M-tiles -> B operand read exactly once).
    const int Ntiles = NPIX / 16;               // 32768 waves per layer
    const int gblocks = Ntiles / 8;             // 8 waves per block
    gemm_bn_relu<8,   32, 0><<<gblocks, 256, 0, stream>>>(x0, w0, b0, g0, t0, m0, v0, x1, nullptr);
    gemm_bn_relu<32,  64, 0><<<gblocks, 256, 0, stream>>>(x1, w1, b1, g1, t1, m1, v1, x2, nullptr);
    zero_kernel<<<(BB*128*SS)/256, 256, 0, stream>>>(out_pts, BB*128*SS);
    gemm_bn_relu<128,128, 1><<<gblocks, 256, 0, stream>>>(x2, w2, b2, g2, t2, m2, v2, nullptr, out_pts);
}